// GraphCNN_89515708383972
// MI455X (gfx1250) — compile-verified
//
#include <hip/hip_runtime.h>
#include <hip/hip_bf16.h>

// ---------------------------------------------------------------------------
// GraphCNN forward for MI455X (gfx1250, wave32).
// GEMMs use V_WMMA_F32_16X16X4_F32 (full fp32 precision, matches reference).
// Edge scatter / pooling use fp32 global atomics (L2-resident working set).
// ---------------------------------------------------------------------------

#define NN 20000      // nodes
#define NE 320000     // edges
#define DD 256        // features
#define NG 64         // graphs
#define NC 10         // classes
#define NH 128        // fc1 hidden

typedef __attribute__((ext_vector_type(2))) float v2f;
typedef __attribute__((ext_vector_type(8))) float v8f;

// ------------------------------ utility ------------------------------------
__global__ void k_zero(float* __restrict__ p, int n) {
    int i = blockIdx.x * blockDim.x + threadIdx.x;
    if (i < n) p[i] = 0.0f;
}

// degree (at dst) via atomics; then dis = rsqrt(deg + 1)
__global__ void k_deg(const int* __restrict__ ei, float* __restrict__ deg) {
    int e = blockIdx.x * blockDim.x + threadIdx.x;
    if (e < NE) atomicAdd(&deg[ei[NE + e]], 1.0f);
}
__global__ void k_dis(float* __restrict__ dis) {
    int i = blockIdx.x * blockDim.x + threadIdx.x;
    if (i < NN) dis[i] = rsqrtf(dis[i] + 1.0f);
}

// ------------------------- GEMM: Out = H @ W  (WMMA fp32) ------------------
// grid = (NN/16, 2), block = 256 (8 waves). Wave w handles tileN = by*8 + w.
// Each wave accumulates one 16x16 f32 tile over K=256 in steps of 4.
__global__ void __launch_bounds__(256)
k_gemm(const float* __restrict__ H, const float* __restrict__ W,
       float* __restrict__ Out) {
    const int lane  = threadIdx.x & 31;
    const int wave  = threadIdx.x >> 5;
    const int tileM = blockIdx.x;
    const int tileN = blockIdx.y * 8 + wave;
    const int half  = lane >> 4;       // 0: K pair (k,k+1), 1: (k+2,k+3)
    const int l16   = lane & 15;
    const int col   = tileN * 16 + l16;

    const float* __restrict__ hrow = H + (tileM * 16 + l16) * DD;

    v8f c = {0.f, 0.f, 0.f, 0.f, 0.f, 0.f, 0.f, 0.f};

#pragma unroll 8
    for (int k = 0; k < DD; k += 4) {
        const int ka = k + half * 2;
        v2f a = *(const v2f*)(hrow + ka);          // A[row][ka], A[row][ka+1]
        v2f b;
        b.x = W[ka * DD + col];                    // B[ka][col]
        b.y = W[(ka + 1) * DD + col];              // B[ka+1][col]
        c = __builtin_amdgcn_wmma_f32_16x16x4_f32(
                /*neg_a=*/false, a, /*neg_b=*/false, b,
                /*c_mod=*/(short)0, c, /*reuse_a=*/false, /*reuse_b=*/false);
    }

    // C/D layout: VGPR v -> M = v + half*8 (lanes 0-15: M=v, 16-31: M=v+8), N = l16
    float* __restrict__ orow = Out + (tileM * 16 + half * 8) * DD + col;
#pragma unroll
    for (int v = 0; v < 8; ++v) orow[v * DD] = c[v];
}

// ------------------- agg init: A = hW * dis^2 + bias -----------------------
__global__ void k_initagg(const float* __restrict__ hw, const float* __restrict__ dis,
                          const float* __restrict__ bias, float* __restrict__ agg) {
    int idx = blockIdx.x * blockDim.x + threadIdx.x;   // NN * 64
    int n = idx >> 6, f = (idx & 63) << 2;
    if (n >= NN) return;
    float s = dis[n]; s = s * s;
    float4 v = *(const float4*)(hw + n * DD + f);
    float4 b = *(const float4*)(bias + f);
    float4 o;
    o.x = v.x * s + b.x; o.y = v.y * s + b.y;
    o.z = v.z * s + b.z; o.w = v.w * s + b.w;
    *(float4*)(agg + n * DD + f) = o;
}

// -------- edge scatter: agg[dst] += hW[src] * dis[src]*dis[dst] ------------
__global__ void k_scatter(const int* __restrict__ ei, const float* __restrict__ hw,
                          const float* __restrict__ dis, float* __restrict__ agg) {
    int idx = blockIdx.x * blockDim.x + threadIdx.x;   // NE * 64
    int e = idx >> 6, f = (idx & 63) << 2;
    if (e >= NE) return;
    int src = ei[e], dst = ei[NE + e];
    float nrm = dis[src] * dis[dst];
    float4 v = *(const float4*)(hw + src * DD + f);
    float* o = agg + dst * DD + f;
    atomicAdd(o + 0, v.x * nrm);
    atomicAdd(o + 1, v.y * nrm);
    atomicAdd(o + 2, v.z * nrm);
    atomicAdd(o + 3, v.w * nrm);
}

// ---------------------- batchnorm statistics -------------------------------
#define BN_ROWS 125   // 160 blocks * 125 rows = 20000
__global__ void k_bnstats(const float* __restrict__ agg, float* __restrict__ gsum) {
    int d = threadIdx.x;                                // 256 features
    int r0 = blockIdx.x * BN_ROWS;
    int r1 = r0 + BN_ROWS; if (r1 > NN) r1 = NN;
    float s = 0.f, q = 0.f;
    for (int r = r0; r < r1; ++r) {
        float v = agg[r * DD + d];
        s += v; q += v * v;
    }
    atomicAdd(&gsum[d], s);
    atomicAdd(&gsum[DD + d], q);
}
__global__ void k_bnfin(const float* __restrict__ gsum, float* __restrict__ mean,
                        float* __restrict__ invstd) {
    int d = threadIdx.x;
    const float invN = 1.0f / (float)NN;
    float m = gsum[d] * invN;
    float v = gsum[DD + d] * invN - m * m;
    mean[d] = m;
    invstd[d] = rsqrtf(fmaxf(v, 0.0f) + 1e-5f);
}
__global__ void k_bnapply(float* __restrict__ h, const float* __restrict__ mean,
                          const float* __restrict__ invstd,
                          const float* __restrict__ gamma,
                          const float* __restrict__ beta) {
    int idx = blockIdx.x * blockDim.x + threadIdx.x;   // NN * 64
    int n = idx >> 6, f = (idx & 63) << 2;
    if (n >= NN) return;
    float4 v  = *(const float4*)(h + n * DD + f);
    float4 m  = *(const float4*)(mean + f);
    float4 is = *(const float4*)(invstd + f);
    float4 g  = *(const float4*)(gamma + f);
    float4 b  = *(const float4*)(beta + f);
    float4 o;
    o.x = fmaxf(g.x * (v.x - m.x) * is.x + b.x, 0.0f);
    o.y = fmaxf(g.y * (v.y - m.y) * is.y + b.y, 0.0f);
    o.z = fmaxf(g.z * (v.z - m.z) * is.z + b.z, 0.0f);
    o.w = fmaxf(g.w * (v.w - m.w) * is.w + b.w, 0.0f);
    *(float4*)(h + n * DD + f) = o;
}

// ------------------------------- pooling -----------------------------------
__global__ void k_counts(const int* __restrict__ batch, float* __restrict__ counts) {
    int n = blockIdx.x * blockDim.x + threadIdx.x;
    if (n < NN) atomicAdd(&counts[batch[n]], 1.0f);
}
__global__ void k_psum(const float* __restrict__ h, const int* __restrict__ batch,
                       float* __restrict__ sums) {
    int idx = blockIdx.x * blockDim.x + threadIdx.x;   // NN * 64
    int n = idx >> 6, f = (idx & 63) << 2;
    if (n >= NN) return;
    int g = batch[n];
    float4 v = *(const float4*)(h + n * DD + f);
    float* o = sums + g * DD + f;
    atomicAdd(o + 0, v.x);
    atomicAdd(o + 1, v.y);
    atomicAdd(o + 2, v.z);
    atomicAdd(o + 3, v.w);
}
__global__ void k_pdiv(float* __restrict__ sums, const float* __restrict__ counts) {
    int idx = blockIdx.x * blockDim.x + threadIdx.x;   // NG * DD
    int g = idx >> 8;
    sums[idx] *= (1.0f / fmaxf(counts[g], 1.0f));
}

// ------------------------------ classifier ---------------------------------
__global__ void k_fc1(const float* __restrict__ pooled, const float* __restrict__ w,
                      const float* __restrict__ b, float* __restrict__ out1) {
    int g = blockIdx.x, j = threadIdx.x;               // 64 x 128
    float acc = b[j];
    const float* p = pooled + g * DD;
    for (int d = 0; d < DD; ++d) acc += p[d] * w[d * NH + j];
    out1[g * NH + j] = fmaxf(acc, 0.0f);
}
__global__ void k_fc2(const float* __restrict__ in, const float* __restrict__ w,
                      const float* __restrict__ b, float* __restrict__ out) {
    int g = blockIdx.x, c = threadIdx.x;               // 64 x 32 (10 active)
    if (c >= NC) return;
    float acc = b[c];
    const float* p = in + g * NH;
    for (int j = 0; j < NH; ++j) acc += p[j] * w[j * NC + c];
    out[g * NC + c] = acc;
}

// ------------------------------ launcher -----------------------------------
extern "C" void kernel_launch(void* const* d_in, const int* in_sizes, int n_in,
                              void* d_out, int out_size, void* d_ws, size_t ws_size,
                              hipStream_t stream) {
    const float* x      = (const float*)d_in[0];
    const int*   ei     = (const int*)  d_in[1];
    const int*   batch  = (const int*)  d_in[2];
    const float* conv_w = (const float*)d_in[3];
    const float* conv_b = (const float*)d_in[4];
    const float* bn_g   = (const float*)d_in[5];
    const float* bn_b   = (const float*)d_in[6];
    const float* fc1w   = (const float*)d_in[7];
    const float* fc1b   = (const float*)d_in[8];
    const float* fc2w   = (const float*)d_in[9];
    const float* fc2b   = (const float*)d_in[10];
    float* out = (float*)d_out;

    // workspace layout (floats)
    float* ws     = (float*)d_ws;
    float* A      = ws;                        // agg / h      (NN*DD = 5,120,000)
    float* B      = ws + 5120000;              // hW           (NN*DD)
    float* dis    = ws + 10240000;             // NN
    float* gsum   = ws + 10260000;             // 512 (sum | sumsq)
    float* mean   = ws + 10260512;             // 256
    float* invstd = ws + 10260768;             // 256
    float* sums   = ws + 10261024;             // NG*DD = 16384
    float* counts = ws + 10277408;             // NG (contiguous after sums)
    float* out1   = ws + 10277472;             // NG*NH = 8192

    // degree -> dis = rsqrt(deg+1)
    k_zero<<<(NN + 255) / 256, 256, 0, stream>>>(dis, NN);
    k_deg <<<(NE + 255) / 256, 256, 0, stream>>>(ei, dis);
    k_dis <<<(NN + 255) / 256, 256, 0, stream>>>(dis);

    const float* h = x;
    for (int L = 0; L < 3; ++L) {
        k_gemm   <<<dim3(NN / 16, 2), 256, 0, stream>>>(h, conv_w + L * DD * DD, B);
        k_initagg<<<(NN * 64) / 256, 256, 0, stream>>>(B, dis, conv_b + L * DD, A);
        k_scatter<<<(NE * 64) / 256, 256, 0, stream>>>(ei, B, dis, A);
        k_zero   <<<2, 256, 0, stream>>>(gsum, 2 * DD);
        k_bnstats<<<160, 256, 0, stream>>>(A, gsum);
        k_bnfin  <<<1, 256, 0, stream>>>(gsum, mean, invstd);
        k_bnapply<<<(NN * 64) / 256, 256, 0, stream>>>(A, mean, invstd,
                                                       bn_g + L * DD, bn_b + L * DD);
        h = A;
    }

    // global mean pool
    k_zero  <<<(NG * DD + NG + 255) / 256, 256, 0, stream>>>(sums, NG * DD + NG);
    k_counts<<<(NN + 255) / 256, 256, 0, stream>>>(batch, counts);
    k_psum  <<<(NN * 64) / 256, 256, 0, stream>>>(A, batch, sums);
    k_pdiv  <<<NG, 256, 0, stream>>>(sums, counts);

    // classifier head
    k_fc1<<<NG, NH, 0, stream>>>(sums, fc1w, fc1b, out1);
    k_fc2<<<NG, 32, 0, stream>>>(out1, fc2w, fc2b, out);
}